// ContinuousLocationMap_62139586839054
// MI455X (gfx1250) — compile-verified
//
#include <hip/hip_runtime.h>
#include <hip/hip_bf16.h>
#include <stdint.h>

// ---- problem constants (match the reference) ----
#define Gdim   2045
#define GG     (2045u * 2045u)       // cells per batch image
#define NBATCH 8
#define NLOCS  1024
#define CHUNK  2048                  // cells staged in LDS per block (32 KB)
#define CORR_BASE 0.634f

// =====================================================================
// Phase 1: fill the base map.
// Each block builds CHUNK cells of the (batch-independent) base pattern in
// LDS, then replicates that tile to all 8 batch copies using the CDNA5
// async LDS->global store path (ASYNCcnt), 16 B per lane per instruction,
// with a non-temporal store hint for the 535 MB write-once stream.
// =====================================================================
__global__ void clm_fill_kernel(float* __restrict__ out) {
#pragma clang fp contract(off)                      // match jnp: mul THEN add
    __shared__ float4 tile[CHUNK];                  // 32 KB
    const unsigned base = blockIdx.x * (unsigned)CHUNK;
    const float DM = (float)(510.5 / 2045.0);       // DELTA_MAP rounded to f32

    // Build tile in LDS: cell f -> (r = f/G, c = f%G),
    // value = {0.634, 0.634, 0.5 + DM*c, 0.5 + DM*r}.
#pragma unroll
    for (int k = 0; k < CHUNK / 256; ++k) {
        unsigned i = threadIdx.x + 256u * k;
        unsigned f = base + i;
        if (f < GG) {
            unsigned r = f / 2045u;
            unsigned c = f - r * 2045u;
            float xc = 0.5f + DM * (float)c;        // contract(off) keeps 2 ops
            float xr = 0.5f + DM * (float)r;
            tile[i] = make_float4(CORR_BASE, CORR_BASE, xc, xr);
        }
    }
    __syncthreads();

    // Stream the tile to the 8 batch copies. SADDR = per-batch base (SGPR
    // pair, uniform), VADDR = 32-bit byte offset within the batch image
    // (max 66.9 MB < 2^31), VSRC = LDS byte address (low 32 bits of the
    // generic pointer are the LDS offset on gfx1250).
    for (int b = 0; b < NBATCH; ++b) {
        unsigned long long bb =
            (unsigned long long)(out + (size_t)b * GG * 4u);
#pragma unroll
        for (int k = 0; k < CHUNK / 256; ++k) {
            unsigned i = threadIdx.x + 256u * k;
            unsigned f = base + i;
            if (f < GG) {
                unsigned dstOff = f * 16u;
                unsigned ldsA   = (unsigned)(unsigned long long)&tile[i];
                asm volatile(
                    "global_store_async_from_lds_b128 %0, %1, %2 th:TH_STORE_NT"
                    :
                    : "v"(dstOff), "v"(ldsA), "s"(bb)
                    : "memory");
            }
        }
    }
    // Ensure all async LDS reads/stores are retired before LDS is released.
    asm volatile("s_wait_asynccnt 0" ::: "memory");
}

// =====================================================================
// Phase 2: scatter with last-write-wins.
// One block per batch. Stage all 1024 window indices in LDS; location l
// owns cell p iff no later location j>l covers p (j covers p <=> idx_j is
// in {p, p+1} on both axes). Winners write {1,1,x,y} — race-free since
// exactly one location per touched cell is uncovered.
// =====================================================================
__global__ void clm_scatter_kernel(const float* __restrict__ locs,
                                   float* __restrict__ out) {
    __shared__ int   sx[NLOCS], sy[NLOCS];
    __shared__ float fx[NLOCS], fy[NLOCS];
    const int b = blockIdx.x;
    const float* lb = locs + (size_t)b * NLOCS * 2;

    for (int l = threadIdx.x; l < NLOCS; l += blockDim.x) {
        float x = lb[2 * l + 0];
        float y = lb[2 * l + 1];
        fx[l] = x;  fy[l] = y;
        sx[l] = (int)(x * 4.0f);    // == trunc(x / LOC_DELTA), exact (0.25)
        sy[l] = (int)(y * 4.0f);
    }
    __syncthreads();

    float* ob = out + (size_t)b * GG * 4u;
    for (int l = threadIdx.x; l < NLOCS; l += blockDim.x) {
        const int ix = sx[l], iy = sy[l];
        bool c00 = false, c01 = false, c10 = false, c11 = false;
        for (int j = l + 1; j < NLOCS; ++j) {
            int jx = sx[j], jy = sy[j];
            bool ax0 = (jx == ix - 1) | (jx == ix);       // covers px = ix-1
            bool ax1 = (jx == ix)     | (jx == ix + 1);   // covers px = ix
            bool ay0 = (jy == iy - 1) | (jy == iy);
            bool ay1 = (jy == iy)     | (jy == iy + 1);
            c00 |= ax0 & ay0;  c01 |= ax0 & ay1;
            c10 |= ax1 & ay0;  c11 |= ax1 & ay1;
            if (c00 & c01 & c10 & c11) break;             // fully overwritten
        }
        float4 v = make_float4(1.0f, 1.0f, fx[l], fy[l]);
        auto wr = [&](int px, int py, bool covered) {
            if (!covered && px >= 0 && px < Gdim && py >= 0 && py < Gdim) {
                *(float4*)(ob + ((size_t)px * Gdim + py) * 4u) = v;
            }
        };
        wr(ix - 1, iy - 1, c00);
        wr(ix - 1, iy,     c01);
        wr(ix,     iy - 1, c10);
        wr(ix,     iy,     c11);
    }
}

extern "C" void kernel_launch(void* const* d_in, const int* in_sizes, int n_in,
                              void* d_out, int out_size, void* d_ws, size_t ws_size,
                              hipStream_t stream) {
    (void)in_sizes; (void)n_in; (void)d_ws; (void)ws_size; (void)out_size;
    const float* locs = (const float*)d_in[0];     // [8, 1024, 2] f32
    float* out = (float*)d_out;                    // [8, 2045, 2045, 4] f32

    const unsigned nblk = (GG + CHUNK - 1) / CHUNK;   // 2043
    clm_fill_kernel<<<nblk, 256, 0, stream>>>(out);
    // Same stream: scatter is ordered after the fill, so its writes win.
    clm_scatter_kernel<<<NBATCH, 256, 0, stream>>>(locs, out);
}